// Deepseekv3Gate_206158430270
// MI455X (gfx1250) — compile-verified
//
#include <hip/hip_runtime.h>
#include <math.h>

#define TOP_K 8
#define N_GROUP 8
#define TOPK_GROUP 4
#define NUM_EXPERTS 256
#define HIDDEN 7168
#define GS (NUM_EXPERTS / N_GROUP) /* 32 experts per group */
#define ROUTED_SCALE 2.5f

#define MTILE 32         /* tokens per block (2 WMMA M-tiles) */
#define KC 64            /* K-chunk staged in LDS */
#define LDA (KC + 4)     /* 16B-aligned rows, banks spread by 4 per row */
#define BLOCK_THREADS 512

typedef __attribute__((ext_vector_type(2))) float v2f;
typedef __attribute__((ext_vector_type(8))) float v8f;

__device__ __forceinline__ float sigmoidf_fast(float x) {
    return 1.0f / (1.0f + __expf(-x));
}

__global__ __launch_bounds__(BLOCK_THREADS)
void Deepseekv3Gate_kernel(const float* __restrict__ hidden,
                           const float* __restrict__ weight,
                           const float* __restrict__ bias,
                           float* __restrict__ out)
{
    __shared__ float sA[MTILE * LDA];                 // 32 x KC A-chunk (padded)
    __shared__ float sLogits[MTILE * NUM_EXPERTS];    // 32 tokens x 256 experts
    __shared__ float sBias[NUM_EXPERTS];
    __shared__ unsigned sMask[MTILE][N_GROUP];        // 256-bit select mask / token
    __shared__ float sDenom[MTILE];

    const int tid    = threadIdx.x;
    const int wave   = tid >> 5;             // 0..15
    const int lane   = tid & 31;
    const int m_base = blockIdx.x * MTILE;   // token tile (32 tokens)
    const int e_base = wave * 16;            // this wave's 16 experts

    if (tid < NUM_EXPERTS) sBias[tid] = bias[tid];

    // WMMA f32 16x16x4 operand lane mapping:
    //  A: lane l holds A[M = l&15][k + 2*(l>>4) + {0,1}] in 2 VGPRs
    //  B: lane l holds B[k + 2*(l>>4) + {0,1}][N = l&15] == W[e_base + (l&15)][k..]
    const int mn  = lane & 15;               // M row (A) / N col (B)
    const int kph = (lane >> 4) * 2;         // K phase: 0 or 2

    // A-stage mapping: one float4 per thread per chunk (32 rows x 16 float4s)
    const int ldr = tid >> 4;                // row 0..31
    const int ldc = (tid & 15) * 4;          // col 0..60 step 4

    v8f acc0 = {};                           // tokens [m_base,      m_base+16)
    v8f acc1 = {};                           // tokens [m_base+16,   m_base+32)
    const float* wrow = weight + (size_t)(e_base + mn) * HIDDEN;

    for (int k0 = 0; k0 < HIDDEN; k0 += KC) {
        __syncthreads();  // protect sA from previous iteration's readers
        // cooperative stage of A chunk: 32 rows x KC cols, one float4/thread
        {
            const float4 av = *(const float4*)(hidden +
                (size_t)(m_base + ldr) * HIDDEN + (k0 + ldc));
            *(float4*)(&sA[ldr * LDA + ldc]) = av;
        }
        __syncthreads();

#pragma unroll
        for (int kk = 0; kk < KC; kk += 4) {
            v2f a0, a1, b;
            const float2 bv = *(const float2*)(wrow + k0 + kk + kph);
            b.x = bv.x;
            b.y = bv.y;
            a0.x = sA[mn * LDA + kk + kph];
            a0.y = sA[mn * LDA + kk + kph + 1];
            a1.x = sA[(mn + 16) * LDA + kk + kph];
            a1.y = sA[(mn + 16) * LDA + kk + kph + 1];
            // Two independent accumulator chains sharing one B operand
            acc0 = __builtin_amdgcn_wmma_f32_16x16x4_f32(
                false, a0, false, b, (short)0, acc0, false, false);
            acc1 = __builtin_amdgcn_wmma_f32_16x16x4_f32(
                false, a1, false, b, (short)0, acc1, false, false);
        }
    }

    // Dump both 16x16 logit tiles into LDS.
    // D layout: VGPR r -> lanes 0-15: M=r, N=lane; lanes 16-31: M=r+8, N=lane-16
    {
        const int mofs = (lane >> 4) * 8;
#pragma unroll
        for (int r = 0; r < 8; ++r) {
            sLogits[(mofs + r) * NUM_EXPERTS + (e_base + mn)] = acc0[r];
            sLogits[(16 + mofs + r) * NUM_EXPERTS + (e_base + mn)] = acc1[r];
        }
    }
    __syncthreads();

    // --- Routing: one thread per token (32 threads), exact top-k semantics ---
    if (tid < MTILE) {
        const int t = tid;
        const float* lg = &sLogits[t * NUM_EXPERTS];

        // group score = sum of top-2 biased scores within each 32-expert group
        float gsc[N_GROUP];
#pragma unroll
        for (int g = 0; g < N_GROUP; ++g) {
            float m1 = -INFINITY, m2 = -INFINITY;
            for (int j = 0; j < GS; ++j) {
                const int e = g * GS + j;
                const float s = sigmoidf_fast(lg[e]) + sBias[e];
                if (s > m1)      { m2 = m1; m1 = s; }
                else if (s > m2) { m2 = s; }
            }
            gsc[g] = m1 + m2;
        }

        // top TOPK_GROUP groups (strict > argmax == lowest-index tie-break)
        unsigned gmask = 0;
        for (int i = 0; i < TOPK_GROUP; ++i) {
            int best = 0; float bv = -INFINITY;
            for (int g = 0; g < N_GROUP; ++g) {
                if (!((gmask >> g) & 1u) && gsc[g] > bv) { bv = gsc[g]; best = g; }
            }
            gmask |= 1u << best;
        }

        // top TOP_K experts over masked biased scores (masked-out groups read 0.0,
        // matching reference swb * mask before top_k)
        unsigned emask[N_GROUP] = {0, 0, 0, 0, 0, 0, 0, 0};
        float denom = 0.0f;
        for (int i = 0; i < TOP_K; ++i) {
            int bi = 0; float bv = -INFINITY;
            for (int e = 0; e < NUM_EXPERTS; ++e) {
                const int g = e >> 5;
                if ((emask[g] >> (e & 31)) & 1u) continue;  // already taken
                float v;
                if ((gmask >> g) & 1u)
                    v = sigmoidf_fast(lg[e]) + sBias[e];
                else
                    v = 0.0f;                               // masked entry
                if (v > bv) { bv = v; bi = e; }
            }
            emask[bi >> 5] |= 1u << (bi & 31);
            denom += sigmoidf_fast(lg[bi]);                 // raw sigmoid score
        }

#pragma unroll
        for (int g = 0; g < N_GROUP; ++g) sMask[t][g] = emask[g];
        sDenom[t] = denom + 1e-20f;
    }
    __syncthreads();

    // --- Coalesced output write: sel / denom * 2.5, zeros elsewhere ---
    for (int i = tid; i < MTILE * NUM_EXPERTS; i += BLOCK_THREADS) {
        const int t = i >> 8;
        const int e = i & (NUM_EXPERTS - 1);
        float v = 0.0f;
        if ((sMask[t][e >> 5] >> (e & 31)) & 1u) {
            v = sigmoidf_fast(sLogits[t * NUM_EXPERTS + e]) / sDenom[t] * ROUTED_SCALE;
        }
        out[(size_t)(m_base + t) * NUM_EXPERTS + e] = v;
    }
}

extern "C" void kernel_launch(void* const* d_in, const int* in_sizes, int n_in,
                              void* d_out, int out_size, void* d_ws, size_t ws_size,
                              hipStream_t stream) {
    const float* hidden = (const float*)d_in[0];   // [T, 7168] fp32
    const float* weight = (const float*)d_in[1];   // [256, 7168] fp32
    const float* bias   = (const float*)d_in[2];   // [256] fp32
    float* out          = (float*)d_out;           // [T, 256] fp32

    const int T = in_sizes[0] / HIDDEN;            // 16384
    const int grid = T / MTILE;                    // 512 blocks, 32 tokens each

    Deepseekv3Gate_kernel<<<grid, BLOCK_THREADS, 0, stream>>>(hidden, weight, bias, out);
}